// TNLayer_79912161509908
// MI455X (gfx1250) — compile-verified
//
#include <hip/hip_runtime.h>
#include <cstdint>

typedef __attribute__((ext_vector_type(2))) float v2f;
typedef __attribute__((ext_vector_type(8))) float v8f;

#define I1D 4
#define I2D 30
#define I3D 30
#define RB  5      // bond dim
#define OD  2      // output dim
#define KDIM (I1D * I2D * I3D)   // 3600
#define BATCH 65536

// ---------------------------------------------------------------------------
// Kernel 1: collapse the TT cores a,b,c,d into Wq (batch-independent).
// Wq element (k, o) stored at float offset (k>>1)*4 + o*2 + (k&1), so that a
// single b64 load per lane yields the (K even, K odd) pair the WMMA B-operand
// VGPR layout wants.  7200 floats = 28.8 KB in d_ws.
// ---------------------------------------------------------------------------
__global__ __launch_bounds__(256)
void tn_build_w(const float* __restrict__ a, const float* __restrict__ b,
                const float* __restrict__ c, const float* __restrict__ d,
                float* __restrict__ wq) {
  int p = blockIdx.x * blockDim.x + threadIdx.x;
  if (p >= KDIM) return;
  int i  = p / (I2D * I3D);
  int j  = (p / I3D) % I2D;
  int k3 = p % I3D;

  // e[s] = sum_r a[i,r] * b[r,j,s]
  float e[RB];
#pragma unroll
  for (int s = 0; s < RB; ++s) {
    float acc = 0.f;
#pragma unroll
    for (int r = 0; r < RB; ++r)
      acc += a[i * RB + r] * b[(r * I2D + j) * RB + s];
    e[s] = acc;
  }
#pragma unroll
  for (int o = 0; o < OD; ++o) {
    float w = 0.f;
#pragma unroll
    for (int t = 0; t < RB; ++t) {
      float f = 0.f;
#pragma unroll
      for (int s = 0; s < RB; ++s)
        f += e[s] * c[(s * OD + o) * RB + t];
      w += f * d[t * I3D + k3];
    }
    wq[(p >> 1) * 4 + o * 2 + (p & 1)] = w;
  }
}

// ---------------------------------------------------------------------------
// Kernel 2: out[b, o] = swish( sum_k x[b,k] * W[o,k] + bias[o] )
// via V_WMMA_F32_16X16X4_F32:  D(16x16) = A(16 rows of x, K=4) * B(4, 16 "o")
// K-loop over 3600 in chunks of 4, two accumulator chains for ILP.
// Each wave handles a 16-row batch tile; block = 8 waves = 128 rows.
// ---------------------------------------------------------------------------
__global__ __launch_bounds__(256)
void tn_wmma_gemv(const float* __restrict__ x, const float* __restrict__ wq,
                  const float* __restrict__ bias, float* __restrict__ out) {
  const int lane = threadIdx.x & 31;
  const int wave = threadIdx.x >> 5;
  const int half = lane >> 4;          // 0: lanes 0-15, 1: lanes 16-31
  const int nl   = lane & 15;          // N index (o) and row-within-tile index
  const int rowbase = blockIdx.x * 128 + wave * 16;

  // A operand: lane holds x[rowbase + nl][k0 + 2*half + {0,1}]  -> one b64 load
  const float* aptr = x + (size_t)(rowbase + nl) * KDIM + 2 * half;

  // B operand: lane holds Wq pair for (k = k0 + 2*half + {0,1}, o = nl).
  // Lanes with nl >= 2 are the zero-padding of N: give them a safe in-range
  // address (wq + k0*2) and cndmask the value to 0 -- no EXEC divergence,
  // so EXEC stays all-ones for the WMMAs.
  const bool bvalid = (nl < OD);
  const float* bptr = wq + (bvalid ? (half * 4 + nl * 2) : 0);
  const v2f bzero = {0.f, 0.f};

  v8f acc0 = {};
  v8f acc1 = {};

#pragma unroll 4
  for (int k0 = 0; k0 < KDIM; k0 += 8) {
    v2f av0 = *(const v2f*)(aptr + k0);
    v2f av1 = *(const v2f*)(aptr + k0 + 4);
    v2f bv0 = *(const v2f*)(bptr + k0 * 2);
    v2f bv1 = *(const v2f*)(bptr + k0 * 2 + 8);
    bv0 = bvalid ? bv0 : bzero;
    bv1 = bvalid ? bv1 : bzero;
    // (neg_a, A, neg_b, B, c_mod, C, reuse_a, reuse_b)
    acc0 = __builtin_amdgcn_wmma_f32_16x16x4_f32(
        false, av0, false, bv0, (short)0, acc0, false, false);
    acc1 = __builtin_amdgcn_wmma_f32_16x16x4_f32(
        false, av1, false, bv1, (short)0, acc1, false, false);
  }

  // D layout: lane (half, nl), VGPR r  <->  row = rowbase + r + 8*half, o = nl
  if (nl < OD) {
    const float bo = bias[nl];
#pragma unroll
    for (int r = 0; r < 8; ++r) {
      const int row = rowbase + r + 8 * half;
      float y = acc0[r] + acc1[r] + bo;
      out[row * OD + nl] = y / (1.f + __expf(-y));   // swish
    }
  }
}

// ---------------------------------------------------------------------------
extern "C" void kernel_launch(void* const* d_in, const int* in_sizes, int n_in,
                              void* d_out, int out_size, void* d_ws, size_t ws_size,
                              hipStream_t stream) {
  (void)in_sizes; (void)n_in; (void)out_size; (void)ws_size;
  const float* x    = (const float*)d_in[0];
  const float* a    = (const float*)d_in[1];
  const float* b    = (const float*)d_in[2];
  const float* c    = (const float*)d_in[3];
  const float* d    = (const float*)d_in[4];
  const float* bias = (const float*)d_in[5];
  float* out = (float*)d_out;
  float* wq  = (float*)d_ws;   // 7200 floats

  tn_build_w<<<(KDIM + 255) / 256, 256, 0, stream>>>(a, b, c, d, wq);
  tn_wmma_gemv<<<BATCH / 128, 256, 0, stream>>>(x, wq, bias, out);
}